// AttentionBlock_240518168646
// MI455X (gfx1250) — compile-verified
//
#include <hip/hip_runtime.h>
#include <math.h>
#include <stdint.h>

typedef __attribute__((ext_vector_type(2))) float v2f;
typedef __attribute__((ext_vector_type(8))) float v8f;

#define B_ 2
#define C_ 256
#define T_ 4096
#define NH_ 4
#define CH_ 64
#define GROUPS_ 32
#define CPG_ 8    // channels per group
#define VSTR 20   // V-tile LDS row stride (floats): 80B, 16B-aligned, bank-conflict-free

static __device__ __forceinline__ v8f wmma_f32(v2f a, v2f b, v8f c) {
  // D = A(16x4) * B(4x16) + C, fp32
  return __builtin_amdgcn_wmma_f32_16x16x4_f32(
      /*neg_a=*/false, a, /*neg_b=*/false, b,
      /*c_mod=*/(short)0, c, /*reuse_a=*/false, /*reuse_b=*/false);
}

static __device__ __forceinline__ void async_b128_to_lds(unsigned lds_off,
                                                         const void* gaddr) {
  unsigned long long ga = (unsigned long long)(uintptr_t)gaddr;
  asm volatile("global_load_async_to_lds_b128 %0, %1, off"
               :: "v"(lds_off), "v"(ga)
               : "memory");
}

static __device__ __forceinline__ void wait_asynccnt0() {
  asm volatile("s_wait_asynccnt 0x0" ::: "memory");
}

// ---------------------------------------------------------------------------
// Kernel 1: GroupNorm over (B, C, T).  One block per (b, group).
// ---------------------------------------------------------------------------
__global__ void gn_kernel(const float* __restrict__ x,
                          const float* __restrict__ w,
                          const float* __restrict__ bgn,
                          float* __restrict__ xn) {
  const int b = blockIdx.x / GROUPS_;
  const int g = blockIdx.x % GROUPS_;
  const float* xp = x + ((size_t)b * C_ + (size_t)g * CPG_) * T_;
  float* xo = xn + ((size_t)b * C_ + (size_t)g * CPG_) * T_;

  float s = 0.f, s2 = 0.f;
  for (int i = threadIdx.x; i < CPG_ * T_; i += blockDim.x) {
    float v = xp[i];
    s += v;
    s2 += v * v;
  }
  __shared__ float red0[256];
  __shared__ float red1[256];
  red0[threadIdx.x] = s;
  red1[threadIdx.x] = s2;
  __syncthreads();
  for (int off = 128; off > 0; off >>= 1) {
    if ((int)threadIdx.x < off) {
      red0[threadIdx.x] += red0[threadIdx.x + off];
      red1[threadIdx.x] += red1[threadIdx.x + off];
    }
    __syncthreads();
  }
  const float invN = 1.0f / (float)(CPG_ * T_);
  const float mean = red0[0] * invN;
  const float var = red1[0] * invN - mean * mean;
  const float rstd = rsqrtf(var + 1e-5f);

  for (int i = threadIdx.x; i < CPG_ * T_; i += blockDim.x) {
    int c = g * CPG_ + i / T_;
    xo[i] = (xp[i] - mean) * rstd * w[c] + bgn[c];
  }
}

// ---------------------------------------------------------------------------
// Kernel 2: QKV GEMM.  out[b,o,t] = sum_c qkv_w[o,c]*xn[b,c,t] + qkv_b[o]
// One wave -> 16(M=o) x 64(N=t) tile, K=256 in steps of 4 (fp32 WMMA).
// ---------------------------------------------------------------------------
__global__ void qkv_gemm(const float* __restrict__ xn,
                         const float* __restrict__ w,
                         const float* __restrict__ bias,
                         float* __restrict__ out) {
  const int wid = blockIdx.x * (blockDim.x >> 5) + (threadIdx.x >> 5);
  const int lane = threadIdx.x & 31;
  const int lo = lane & 15;
  const int hi = lane >> 4;
  const int kb = hi * 2;

  const int b = wid / (48 * 64);          // 768/16=48 mtiles, 4096/64=64 ntiles
  const int rem = wid % (48 * 64);
  const int m0 = (rem / 64) * 16;
  const int n0 = (rem % 64) * 64;

  const float* xb = xn + (size_t)b * C_ * T_;
  v8f acc[4] = {};

  for (int c = 0; c < C_; c += 4) {
    v2f av = *(const v2f*)(w + (size_t)(m0 + lo) * C_ + c + kb);
#pragma unroll
    for (int nt = 0; nt < 4; ++nt) {
      const float* bp = xb + (size_t)(c + kb) * T_ + n0 + nt * 16 + lo;
      v2f bb;
      bb.x = bp[0];
      bb.y = bp[T_];
      acc[nt] = wmma_f32(av, bb, acc[nt]);
    }
  }

  float* ob = out + (size_t)b * (3 * C_) * T_;
#pragma unroll
  for (int r = 0; r < 8; ++r) {
    const int row = m0 + r + 8 * hi;
    const float bv = bias[row];
#pragma unroll
    for (int nt = 0; nt < 4; ++nt) {
      ob[(size_t)row * T_ + n0 + nt * 16 + lo] = acc[nt][r] + bv;
    }
  }
}

// ---------------------------------------------------------------------------
// Kernel 3: Flash attention per (b*NH) head-batch.
// qkv layout per b: 768 rows; head h owns rows [h*192, h*192+192):
//   q = +0..63, k = +64..127, v = +128..191, each (64, T).
// Block = 4 waves x 16 query rows = 64 queries.  K/V tiles (shared by all
// 4 waves) are double-buffered in LDS via async-to-LDS B128 copies.
// ---------------------------------------------------------------------------
__global__ void flash_attn(const float* __restrict__ qkv,
                           float* __restrict__ aout) {
  __shared__ float ktile[2][64 * 16];     // [c][s], row stride 16
  __shared__ float vtile[2][64 * VSTR];   // [c][s], row stride 20 (padded)
  __shared__ float plds[4][16 * 20];      // per-wave P tile, stride 20

  const int tid = threadIdx.x;            // 0..127
  const int wv = tid >> 5;
  const int lane = tid & 31;
  const int lo = lane & 15;
  const int hi = lane >> 4;
  const int kb = hi * 2;

  const int bh = blockIdx.x >> 6;         // 0..7
  const int qb = blockIdx.x & 63;         // 64 query blocks of 64
  const int t0 = qb * 64 + wv * 16;
  const int b = bh >> 2, h = bh & 3;

  const float* base = qkv + ((size_t)b * (3 * C_) + (size_t)h * 192) * T_;
  const float* qp = base;
  const float* kp = base + (size_t)64 * T_;
  const float* vp = base + (size_t)128 * T_;

  // Cooperative async staging of one K tile + one V tile (64 rows x 16 s).
  // 256 16B-chunks per tile, 128 threads -> 2 chunks/thread per tile.
  auto stage = [&](int bu, int s0) {
#pragma unroll
    for (int cidx = 0; cidx < 2; ++cidx) {
      const int chunk = cidx * 128 + tid;
      const int row = chunk >> 2;
      const int qtr = chunk & 3;
      async_b128_to_lds((unsigned)(uintptr_t)&ktile[bu][row * 16 + qtr * 4],
                        kp + (size_t)row * T_ + s0 + qtr * 4);
      async_b128_to_lds((unsigned)(uintptr_t)&vtile[bu][row * VSTR + qtr * 4],
                        vp + (size_t)row * T_ + s0 + qtr * 4);
    }
  };

  // Q tile (16 queries x 64 channels) in A layout, held for whole key loop
  v2f qa[16];
#pragma unroll
  for (int j = 0; j < 16; ++j) {
    const float* p = qp + (size_t)(4 * j + kb) * T_ + t0 + lo;
    qa[j].x = p[0];
    qa[j].y = p[T_];
  }

  v8f oacc[4] = {};
  float mrow[8], ssum[8];
#pragma unroll
  for (int r = 0; r < 8; ++r) {
    mrow[r] = -1e30f;
    ssum[r] = 0.f;
  }

  stage(0, 0);
  wait_asynccnt0();
  __syncthreads();

  int bu = 0;
  for (int s0 = 0; s0 < T_; s0 += 16) {
    if (s0 + 16 < T_) stage(bu ^ 1, s0 + 16);  // prefetch next tile

    // S = Q^T K tile (16x16), K=64 via 16 WMMA steps; K from LDS
    v8f sacc = {};
#pragma unroll
    for (int j = 0; j < 16; ++j) {
      v2f bb;
      bb.x = ktile[bu][(4 * j + kb) * 16 + lo];
      bb.y = ktile[bu][(4 * j + kb + 1) * 16 + lo];
      sacc = wmma_f32(qa[j], bb, sacc);
    }

    // Online softmax.  Row r+8*hi lives across the 16 lanes of a half-wave.
    float pr[8], alpha[8];
#pragma unroll
    for (int r = 0; r < 8; ++r) {
      float sv = sacc[r] * 0.125f;  // scale^2 = ch^-0.5 = 1/8
      float mx = sv;
      mx = fmaxf(mx, __shfl_xor(mx, 1, 32));
      mx = fmaxf(mx, __shfl_xor(mx, 2, 32));
      mx = fmaxf(mx, __shfl_xor(mx, 4, 32));
      mx = fmaxf(mx, __shfl_xor(mx, 8, 32));
      const float mnew = fmaxf(mrow[r], mx);
      alpha[r] = __expf(mrow[r] - mnew);
      mrow[r] = mnew;
      const float pv = __expf(sv - mnew);
      pr[r] = pv;
      float ps = pv;
      ps += __shfl_xor(ps, 1, 32);
      ps += __shfl_xor(ps, 2, 32);
      ps += __shfl_xor(ps, 4, 32);
      ps += __shfl_xor(ps, 8, 32);
      ssum[r] = ssum[r] * alpha[r] + ps;
    }
#pragma unroll
    for (int ct = 0; ct < 4; ++ct)
#pragma unroll
      for (int r = 0; r < 8; ++r) oacc[ct][r] *= alpha[r];

    // P: D layout -> LDS -> A layout (per-wave region)
#pragma unroll
    for (int r = 0; r < 8; ++r) plds[wv][(r + 8 * hi) * 20 + lo] = pr[r];
    __syncthreads();

    // O += P * V^T : K=16 via 4 WMMA steps, 4 channel tiles; V from LDS
#pragma unroll
    for (int j2 = 0; j2 < 4; ++j2) {
      v2f pa = *(const v2f*)&plds[wv][lo * 20 + j2 * 4 + kb];
#pragma unroll
      for (int ct = 0; ct < 4; ++ct) {
        v2f bb = *(const v2f*)&vtile[bu][(ct * 16 + lo) * VSTR + j2 * 4 + kb];
        oacc[ct] = wmma_f32(pa, bb, oacc[ct]);
      }
    }

    wait_asynccnt0();   // our async writes into bu^1 have landed
    __syncthreads();    // everyone done reading bu / writing bu^1
    bu ^= 1;
  }

  // Normalize and write a[b, h*64 + c, t]
  float* ab = aout + ((size_t)b * C_ + (size_t)h * CH_) * T_;
#pragma unroll
  for (int r = 0; r < 8; ++r) {
    const float inv = 1.0f / ssum[r];
#pragma unroll
    for (int ct = 0; ct < 4; ++ct) {
      ab[(size_t)(ct * 16 + lo) * T_ + t0 + r + 8 * hi] = oacc[ct][r] * inv;
    }
  }
}

// ---------------------------------------------------------------------------
// Kernel 4: proj GEMM + bias + residual -> out
// ---------------------------------------------------------------------------
__global__ void proj_gemm(const float* __restrict__ a,
                          const float* __restrict__ w,
                          const float* __restrict__ bias,
                          const float* __restrict__ x,
                          float* __restrict__ out) {
  const int wid = blockIdx.x * (blockDim.x >> 5) + (threadIdx.x >> 5);
  const int lane = threadIdx.x & 31;
  const int lo = lane & 15;
  const int hi = lane >> 4;
  const int kb = hi * 2;

  const int b = wid / (16 * 64);          // 256/16=16 mtiles, 64 ntiles
  const int rem = wid % (16 * 64);
  const int m0 = (rem / 64) * 16;
  const int n0 = (rem % 64) * 64;

  const float* ab = a + (size_t)b * C_ * T_;
  v8f acc[4] = {};

  for (int c = 0; c < C_; c += 4) {
    v2f av = *(const v2f*)(w + (size_t)(m0 + lo) * C_ + c + kb);
#pragma unroll
    for (int nt = 0; nt < 4; ++nt) {
      const float* bp = ab + (size_t)(c + kb) * T_ + n0 + nt * 16 + lo;
      v2f bb;
      bb.x = bp[0];
      bb.y = bp[T_];
      acc[nt] = wmma_f32(av, bb, acc[nt]);
    }
  }

  const float* xb = x + (size_t)b * C_ * T_;
  float* ob = out + (size_t)b * C_ * T_;
#pragma unroll
  for (int r = 0; r < 8; ++r) {
    const int row = m0 + r + 8 * hi;
    const float bv = bias[row];
#pragma unroll
    for (int nt = 0; nt < 4; ++nt) {
      const size_t idx = (size_t)row * T_ + n0 + nt * 16 + lo;
      ob[idx] = xb[idx] + acc[nt][r] + bv;
    }
  }
}

// ---------------------------------------------------------------------------
extern "C" void kernel_launch(void* const* d_in, const int* in_sizes, int n_in,
                              void* d_out, int out_size, void* d_ws, size_t ws_size,
                              hipStream_t stream) {
  (void)in_sizes; (void)n_in; (void)out_size; (void)ws_size;
  const float* x  = (const float*)d_in[0];
  const float* nw = (const float*)d_in[1];
  const float* nb = (const float*)d_in[2];
  const float* qw = (const float*)d_in[3];
  const float* qb = (const float*)d_in[4];
  const float* pw = (const float*)d_in[5];
  const float* pb = (const float*)d_in[6];
  float* out = (float*)d_out;

  float* ws = (float*)d_ws;
  float* xn   = ws;                                   //  8 MB: (2,256,4096)
  float* qkv  = xn + (size_t)B_ * C_ * T_;            // 24 MB: (2,768,4096)
  float* attn = qkv + (size_t)B_ * 3 * C_ * T_;       //  8 MB: (2,256,4096)

  gn_kernel<<<B_ * GROUPS_, 256, 0, stream>>>(x, nw, nb, xn);
  qkv_gemm<<<(B_ * 48 * 64) / 8, 256, 0, stream>>>(xn, qw, qb, qkv);
  flash_attn<<<B_ * NH_ * (T_ / 64), 128, 0, stream>>>(qkv, attn);
  proj_gemm<<<(B_ * 16 * 64) / 8, 256, 0, stream>>>(attn, pw, pb, x, out);
}